// dice_shape_loss_81389630259398
// MI455X (gfx1250) — compile-verified
//
#include <hip/hip_runtime.h>

// ---- types ------------------------------------------------------------
typedef float v4f __attribute__((ext_vector_type(4)));
typedef float v2f __attribute__((ext_vector_type(2)));
typedef float v8f __attribute__((ext_vector_type(8)));

#define GRID1 1024
#define BLOCK 256
#define LOG_CLAMP -100.0f

// ---------------------------------------------------------------------
// Kernel 1: streaming BCE + shape-weight, one fp32 partial per block.
// Memory-bound: 3x b128 non-temporal loads per 4 elements, native v_log_f32.
// ---------------------------------------------------------------------
__global__ __launch_bounds__(BLOCK) void bce_partial_kernel(
    const float* __restrict__ y_true,
    const float* __restrict__ y_pred,
    const float* __restrict__ shapeloss,
    float* __restrict__ partial,
    int n)
{
    const int nvec = n >> 2;                       // N divisible by 4 here
    const v4f* yt4 = (const v4f*)y_true;
    const v4f* yp4 = (const v4f*)y_pred;
    const v4f* sl4 = (const v4f*)shapeloss;

    float acc = 0.0f;
    const int stride = gridDim.x * blockDim.x;
    for (int i = blockIdx.x * blockDim.x + threadIdx.x; i < nvec; i += stride) {
        v4f t = __builtin_nontemporal_load(yt4 + i);   // global_load_b128, TH=NT
        v4f p = __builtin_nontemporal_load(yp4 + i);
        v4f w = __builtin_nontemporal_load(sl4 + i);
#pragma unroll
        for (int c = 0; c < 4; ++c) {
            // BCE: a = -(t*log(p) + (1-t)*log(1-p)), log terms clamped at -100
            float lp   = fmaxf(__logf(p[c]),        LOG_CLAMP);  // v_log_f32 path
            float l1mp = fmaxf(__logf(1.0f - p[c]), LOG_CLAMP);  // exact 1-p for p>=0.5
            float a    = -(t[c] * lp + (1.0f - t[c]) * l1mp);
            // contribution: a*shapeloss + a
            acc = fmaf(a, w[c], acc + a);
        }
    }

    // generic tail (N % 4), handled by one thread; N here is a multiple of 4
    if (blockIdx.x == 0 && threadIdx.x == 0) {
        for (int j = nvec << 2; j < n; ++j) {
            float lp   = fmaxf(__logf(y_pred[j]),        LOG_CLAMP);
            float l1mp = fmaxf(__logf(1.0f - y_pred[j]), LOG_CLAMP);
            float a    = -(y_true[j] * lp + (1.0f - y_true[j]) * l1mp);
            acc = fmaf(a, shapeloss[j], acc + a);
        }
    }

    // deterministic block tree-reduction in LDS
    __shared__ float red[BLOCK];
    red[threadIdx.x] = acc;
    __syncthreads();
#pragma unroll
    for (int off = BLOCK / 2; off > 0; off >>= 1) {
        if (threadIdx.x < off) red[threadIdx.x] += red[threadIdx.x + off];
        __syncthreads();
    }
    if (threadIdx.x == 0) partial[blockIdx.x] = red[0];
}

// ---------------------------------------------------------------------
// Kernel 2: final reduce of GRID1 partials using V_WMMA_F32_16X16X4_F32
// with a ones B-matrix as the cross-lane summation engine, then *1/N.
// All threads execute the WMMA uniformly (EXEC all-ones per wave).
// ---------------------------------------------------------------------
__global__ __launch_bounds__(BLOCK) void bce_final_kernel(
    const float* __restrict__ partial,
    float* __restrict__ out,
    float invN)
{
    __shared__ float lds[BLOCK];

    // 1024 partials -> 256 per-thread sums
    float v = 0.0f;
#pragma unroll
    for (int k = 0; k < GRID1 / BLOCK; ++k)
        v += partial[threadIdx.x + k * BLOCK];
    lds[threadIdx.x] = v;
    __syncthreads();

    // every wave reduces the same 256 values: lane L sums lds[8L..8L+7]
    const int lane = threadIdx.x & 31;
    float s = 0.0f;
#pragma unroll
    for (int j = 0; j < 8; ++j)
        s += lds[lane * 8 + j];

    // A (16x4 f32, 2 VGPRs): VGPR0 = s per lane, VGPR1 = 0
    //   => row m of A = { s[m], 0, s[m+16], 0 }
    // B (4x16) = all ones  => D[m][n] = s[m] + s[m+16]
    v2f A;  A.x = s;    A.y = 0.0f;
    v2f Bm; Bm.x = 1.0f; Bm.y = 1.0f;
    v8f C = {};
    v8f D = __builtin_amdgcn_wmma_f32_16x16x4_f32(
        /*neg_a=*/false, A, /*neg_b=*/false, Bm,
        /*c_mod=*/(short)0, C, /*reuse_a=*/false, /*reuse_b=*/false);

    // lane (<16): sum_r D[r] = sum_{m=0..7}(s[m]+s[m+16])
    // lane (>=16): sum_r D[r] = sum_{m=8..15}(s[m]+s[m+16])
    float ssum = D[0] + D[1] + D[2] + D[3] + D[4] + D[5] + D[6] + D[7];
    float total = __shfl(ssum, 0, 32) + __shfl(ssum, 16, 32);

    if (threadIdx.x == 0) out[0] = total * invN;
}

// ---------------------------------------------------------------------
extern "C" void kernel_launch(void* const* d_in, const int* in_sizes, int n_in,
                              void* d_out, int out_size, void* d_ws, size_t ws_size,
                              hipStream_t stream) {
    const float* y_true    = (const float*)d_in[0];
    const float* y_pred    = (const float*)d_in[1];
    const float* shapeloss = (const float*)d_in[2];
    float* out     = (float*)d_out;
    float* partial = (float*)d_ws;               // needs GRID1*4 = 4 KB scratch
    const int n = in_sizes[0];

    bce_partial_kernel<<<GRID1, BLOCK, 0, stream>>>(y_true, y_pred, shapeloss, partial, n);
    bce_final_kernel<<<1, BLOCK, 0, stream>>>(partial, out, 1.0f / (float)n);
}